// GenerateDetections_1554778161174
// MI455X (gfx1250) — compile-verified
//
#include <hip/hip_runtime.h>
#include <hip/hip_bf16.h>

#define SCORE_THR 0.05f
#define NEG_INF   (-1e9f)
#define MAX_DET   100
#define TILE      1024   // boxes per LDS tile (16 KB of float4); x2 buffers

typedef int   v4i __attribute__((ext_vector_type(4)));
typedef float v4f __attribute__((ext_vector_type(4)));

// ---------------------------------------------------------------------------
// CDNA5 async global->LDS copy (ASYNCcnt-tracked) with portable fallback.
// Builtin signature (from hipcc diagnostic): (global int4*, lds int4*, imm, imm)
// ---------------------------------------------------------------------------
__device__ __forceinline__ void async_copy_b128(const float4* g, float4* l) {
#if defined(__AMDGCN__) && __has_builtin(__builtin_amdgcn_global_load_async_to_lds_b128)
  __builtin_amdgcn_global_load_async_to_lds_b128(
      (__attribute__((address_space(1))) v4i*)g,
      (__attribute__((address_space(3))) v4i*)l,
      0, 0);
#else
  *l = *g;
#endif
}

template <int CNT>
__device__ __forceinline__ void async_wait_le() {
#if defined(__AMDGCN__) && __has_builtin(__builtin_amdgcn_s_wait_asynccnt)
  __builtin_amdgcn_s_wait_asynccnt(CNT);
#elif defined(__gfx1250__)
  asm volatile("s_wait_asynccnt %0" ::"i"(CNT) : "memory");
#endif
}

// ---------------------------------------------------------------------------
// Kernel 1: per-anchor class max + argmax. Bandwidth-dominant phase:
// ~196 MB of score reads -> ~8.4 us floor at 23.3 TB/s.
//  - compile-time C4 => fully unrolled clause of global_load_b128
//  - non-temporal loads: scores are read-once; keep L2 free for the
//    boxes/s working set that the scan phase re-reads 100x.
// ---------------------------------------------------------------------------
template <int C4>
__global__ void class_reduce_fixed_kernel(const v4f* __restrict__ scores4,
                                          float* __restrict__ s_out,
                                          int* __restrict__ cls_out,
                                          int total) {
  int a = blockIdx.x * blockDim.x + threadIdx.x;
  if (a >= total) return;
  const v4f* row = scores4 + (long long)a * C4;
  float m = -3.0e38f;
  int am = 0;
#pragma unroll
  for (int q = 0; q < C4; ++q) {
    v4f v = __builtin_nontemporal_load(row + q);
    const int c = q * 4;
    // strict > keeps first occurrence (jnp.argmax), components in order
    if (v.x > m) { m = v.x; am = c; }
    if (v.y > m) { m = v.y; am = c + 1; }
    if (v.z > m) { m = v.z; am = c + 2; }
    if (v.w > m) { m = v.w; am = c + 3; }
  }
  s_out[a]   = (m > SCORE_THR) ? m : NEG_INF;
  cls_out[a] = am;
}

__global__ void class_reduce_v4_kernel(const v4f* __restrict__ scores4,
                                       float* __restrict__ s_out,
                                       int* __restrict__ cls_out,
                                       int total, int C4) {
  int a = blockIdx.x * blockDim.x + threadIdx.x;
  if (a >= total) return;
  const v4f* row = scores4 + (long long)a * C4;
  float m = -3.0e38f;
  int am = 0;
  for (int q = 0; q < C4; ++q) {
    v4f v = __builtin_nontemporal_load(row + q);
    const int c = q * 4;
    if (v.x > m) { m = v.x; am = c; }
    if (v.y > m) { m = v.y; am = c + 1; }
    if (v.z > m) { m = v.z; am = c + 2; }
    if (v.w > m) { m = v.w; am = c + 3; }
  }
  s_out[a]   = (m > SCORE_THR) ? m : NEG_INF;
  cls_out[a] = am;
}

__global__ void class_reduce_kernel(const float* __restrict__ scores,
                                    float* __restrict__ s_out,
                                    int* __restrict__ cls_out,
                                    int total, int C) {
  int a = blockIdx.x * blockDim.x + threadIdx.x;
  if (a >= total) return;
  const float* row = scores + (long long)a * C;
  float m = row[0];
  int am = 0;
  for (int c = 1; c < C; ++c) {
    float v = row[c];
    if (v > m) { m = v; am = c; }
  }
  s_out[a]   = (m > SCORE_THR) ? m : NEG_INF;
  cls_out[a] = am;
}

// ---------------------------------------------------------------------------
// wave32 / block argmax with first-index tie-break
// ---------------------------------------------------------------------------
__device__ __forceinline__ void wave_argmax(float& v, int& i) {
#pragma unroll
  for (int off = 16; off > 0; off >>= 1) {
    float ov = __shfl_down(v, off, 32);
    int   oi = __shfl_down(i, off, 32);
    if (ov > v || (ov == v && oi < i)) { v = ov; i = oi; }
  }
}

__device__ __forceinline__ void block_argmax(float v, int i,
                                             float* rs, int* ri,
                                             float& out_v, int& out_i) {
  wave_argmax(v, i);
  const int wid  = threadIdx.x >> 5;
  const int lane = threadIdx.x & 31;
  if (lane == 0) { rs[wid] = v; ri[wid] = i; }
  __syncthreads();
  if (wid == 0) {
    const int nw = blockDim.x >> 5;
    v = (lane < nw) ? rs[lane] : -3.0e38f;
    i = (lane < nw) ? ri[lane] : 0x7fffffff;
    wave_argmax(v, i);
    if (lane == 0) { rs[0] = v; ri[0] = i; }
  }
  __syncthreads();
  out_v = rs[0];
  out_i = ri[0];
  __syncthreads();  // rs/ri reusable by next call
}

// ---------------------------------------------------------------------------
// Kernel 2: sequential soft-NMS scan, one workgroup (32 wave32 waves) per
// image. Each of the first 99 of the 100 iterations does ONE fused pass over N:
//   s[n] *= exp(-iou(sel,n)^2 / 0.5); s[sel] = NEG; track running argmax.
// Box tiles are double-buffered through LDS via the CDNA5 async copy engine:
// issue tile t+1, s_wait_asynccnt 1 (async loads complete in order), process
// tile t. Buffer reuse is fenced by the trailing __syncthreads of tile t-1.
// ---------------------------------------------------------------------------
__global__ void __launch_bounds__(1024)
soft_nms_kernel(const float* __restrict__ boxes,
                float* __restrict__ s_ws,
                const int* __restrict__ cls_ws,
                float* __restrict__ out,
                int B, int N) {
  __shared__ float4 tile[2][TILE];
  __shared__ float  rs[32];
  __shared__ int    ri[32];
  __shared__ float  selq[4];
  __shared__ int    s_valid;

  const int b   = blockIdx.x;
  const int tid = threadIdx.x;
  const int nth = blockDim.x;

  const float4* boxB = (const float4*)(boxes + (size_t)b * N * 4);
  float*        sB   = s_ws + (size_t)b * N;
  const int*    clsB = cls_ws + (size_t)b * N;

  float* out_scores = out;                               // [B,100]
  float* out_boxes  = out + (size_t)B * MAX_DET;         // [B,100,4]
  float* out_cls    = out + (size_t)B * MAX_DET * 5;     // [B,100]
  float* out_valid  = out + (size_t)B * MAX_DET * 6;     // [B]

  if (tid == 0) s_valid = 0;

  // Initial argmax over s0 (no decay yet).
  float lv = -3.0e38f;
  int   li = 0x7fffffff;
  for (int n = tid; n < N; n += nth) {
    float v = sB[n];
    if (v > lv) { lv = v; li = n; }  // strided n increases -> first idx kept
  }
  float sc;
  int   idx;
  block_argmax(lv, li, rs, ri, sc, idx);

  const int ntiles = (N + TILE - 1) / TILE;

  for (int k = 0; k < MAX_DET; ++k) {
    const bool ok = sc > SCORE_THR;

    if (tid == 0) {
      out_scores[(size_t)b * MAX_DET + k] = ok ? sc : -1.0f;
      out_cls[(size_t)b * MAX_DET + k]    = ok ? (float)clsB[idx] : -1.0f;
      if (ok) s_valid++;
    }
    if (tid < 4) {
      const int src = ok ? idx : 0;  // TF pad_to_max: padded sel_idx == 0
      out_boxes[((size_t)b * MAX_DET + k) * 4 + tid] =
          ((const float*)&boxB[src])[tid];
      selq[tid] = ((const float*)&boxB[idx])[tid];
    }
    if (k == MAX_DET - 1) break;  // last selection needs no decay pass
    __syncthreads();

    const float qy1 = selq[0], qx1 = selq[1], qy2 = selq[2], qx2 = selq[3];
    const float a1  = (qy2 - qy1) * (qx2 - qx1);

    lv = -3.0e38f;
    li = 0x7fffffff;

    // Prime tile 0 into buffer 0.
    {
      const int cnt0 = min(TILE, N);
      if (tid < cnt0) async_copy_b128(&boxB[tid], &tile[0][tid]);
    }
    for (int t = 0; t < ntiles; ++t) {
      const int base = t * TILE;
      const int cnt  = min(TILE, N - base);
      const int cur  = t & 1;
      if (t + 1 < ntiles) {
        // Issue next tile into the other buffer, then wait only for the
        // older copy (in-order completion) -> copy/compute overlap.
        const int nbase = base + TILE;
        const int ncnt  = min(TILE, N - nbase);
        if (tid < ncnt) async_copy_b128(&boxB[nbase + tid], &tile[cur ^ 1][tid]);
        async_wait_le<1>();
      } else {
        async_wait_le<0>();
      }
      __syncthreads();
      if (tid < cnt) {
        const int n = base + tid;
        float4 bx = tile[cur][tid];
        float yy1 = fmaxf(qy1, bx.x);
        float xx1 = fmaxf(qx1, bx.y);
        float yy2 = fminf(qy2, bx.z);
        float xx2 = fminf(qx2, bx.w);
        float inter = fmaxf(yy2 - yy1, 0.0f) * fmaxf(xx2 - xx1, 0.0f);
        float a2    = (bx.z - bx.x) * (bx.w - bx.y);
        float iou   = inter / fmaxf(a1 + a2 - inter, 1e-12f);
        float w     = __expf(-iou * iou * 2.0f);  // /soft_nms_sigma(=0.5)
        float sv    = sB[n] * w;
        if (n == idx) sv = NEG_INF;               // .at[idx].set(NEG)
        sB[n] = sv;
        if (sv > lv) { lv = sv; li = n; }
      }
      __syncthreads();  // fences buffer (cur) for reuse at t+2
    }
    block_argmax(lv, li, rs, ri, sc, idx);
  }

  if (tid == 0) out_valid[b] = (float)s_valid;
}

// ---------------------------------------------------------------------------
extern "C" void kernel_launch(void* const* d_in, const int* in_sizes, int n_in,
                              void* d_out, int out_size, void* d_ws, size_t ws_size,
                              hipStream_t stream) {
  const float* boxes  = (const float*)d_in[0];  // [B,N,4] f32
  const float* scores = (const float*)d_in[1];  // [B,N,C] f32
  float* out = (float*)d_out;

  // out_size = B*(100 + 400 + 100 + 1) = 601*B
  const int B = out_size / (MAX_DET * 6 + 1);
  const long long BN = (long long)in_sizes[0] / 4;      // B*N
  const int N = (int)(BN / B);
  const int C = (int)((long long)in_sizes[1] / BN);     // 80

  // Workspace layout: s[B*N] f32, cls[B*N] i32  (~4.9 MB total)
  float* s_ws   = (float*)d_ws;
  int*   cls_ws = (int*)(s_ws + BN);

  const int total = (int)BN;
  const int thr = 256;
  const int blocks = (total + thr - 1) / thr;
  if (C == 80) {
    class_reduce_fixed_kernel<20><<<blocks, thr, 0, stream>>>(
        (const v4f*)scores, s_ws, cls_ws, total);
  } else if ((C & 3) == 0) {
    class_reduce_v4_kernel<<<blocks, thr, 0, stream>>>(
        (const v4f*)scores, s_ws, cls_ws, total, C >> 2);
  } else {
    class_reduce_kernel<<<blocks, thr, 0, stream>>>(
        scores, s_ws, cls_ws, total, C);
  }

  soft_nms_kernel<<<B, 1024, 0, stream>>>(boxes, s_ws, cls_ws, out, B, N);
}